// TuckerLayer_69157563400252
// MI455X (gfx1250) — compile-verified
//
#include <hip/hip_runtime.h>

typedef __attribute__((ext_vector_type(2))) float v2f;
typedef __attribute__((ext_vector_type(4))) float f4;
typedef __attribute__((ext_vector_type(8))) float v8f;

// ---------------------------------------------------------------------------
// Front end: v[n,m,i] = ops0[n,m] @ (ops1[n,m] @ mats[n,:,m])
// Writes G_n[i][m] = v[n,m,i]  (row-major 128x128 per n)
// ---------------------------------------------------------------------------
__global__ __launch_bounds__(128)
void front_kernel(const float* __restrict__ ops, const float* __restrict__ mats,
                  float* __restrict__ G)
{
    __shared__ float w[128];
    __shared__ float u[128];
    const int nm = blockIdx.x;           // 0..383
    const int n  = nm >> 7;
    const int m  = nm & 127;
    const int t  = threadIdx.x;

    // w[k] = mats[n,k,m]
    w[t] = mats[(n * 128 + t) * 128 + m];
    __syncthreads();

    const float* ops1 = ops + ((long)nm * 2 + 1) * 16384;
    float s = 0.f;
    #pragma unroll 8
    for (int k = 0; k < 128; ++k) s += ops1[t * 128 + k] * w[k];
    u[t] = s;
    __syncthreads();

    const float* ops0 = ops + (long)nm * 2 * 16384;
    float s2 = 0.f;
    #pragma unroll 8
    for (int j = 0; j < 128; ++j) s2 += ops0[t * 128 + j] * u[j];

    G[n * 16384 + t * 128 + m] = s2;     // G_n[i=t][m]
}

// ---------------------------------------------------------------------------
// Generic f32 WMMA GEMM:  C[M,N] = A[M,K] * op(B)   (op = identity or trans)
// Block = 256 threads = 8 waves. Block tile: 128(M) x 64(N).
// Wave w computes rows [blkM + 16w, +16) x cols [blkN, +64) as 4 16x16 tiles.
// Uses V_WMMA_F32_16X16X4_F32 (full f32 precision, K stepped by 4).
//   A frag layout: a.x = A[m0+l16][kk+2*half], a.y = +1   (ISA 16x4 f32 table)
//   B frag layout: b.x = B[kk+2*half][n],      b.y = row+1
//   D layout: acc*[v] -> C[m0 + v + 8*half][n0 + 16t + l16]
// TRANSB is a template parameter: no control flow inside the K-loop.
// ---------------------------------------------------------------------------
template <bool TRANSB>
__global__ __launch_bounds__(256)
void gemm_wmma_f32(const float* __restrict__ A, long lda, long strideA,
                   const float* __restrict__ B, long ldb, long strideB,
                   float* __restrict__ C, long ldc, long strideC,
                   int M, int N, int K)
{
    const int tid  = threadIdx.x;
    const int wave = tid >> 5;
    const int lane = tid & 31;
    const int half = lane >> 4;      // 0 or 1
    const int l16  = lane & 15;

    const long batch = blockIdx.z;
    A += batch * strideA;
    B += batch * strideB;
    C += batch * strideC;

    const int m0 = blockIdx.y * 128 + wave * 16;
    const int n0 = blockIdx.x * 64;
    if (m0 >= M) return;             // wave-uniform (EXEC stays all-ones)

    v8f acc0 = {}, acc1 = {}, acc2 = {}, acc3 = {};

    const float* aRow = A + (long)(m0 + l16) * lda + 2 * half;  // + kk -> frag
    const int    c    = n0 + l16;

    if (TRANSB) {
        // B logical [k][n] = Bmat[n][k] : contiguous K pairs per lane
        const float* b0p = B + (long)(c +  0) * ldb + 2 * half;
        const float* b1p = B + (long)(c + 16) * ldb + 2 * half;
        const float* b2p = B + (long)(c + 32) * ldb + 2 * half;
        const float* b3p = B + (long)(c + 48) * ldb + 2 * half;
        #pragma unroll 4
        for (int kk = 0; kk < K; kk += 4) {
            const v2f a  = *(const v2f*)(aRow + kk);
            const v2f b0 = *(const v2f*)(b0p + kk);
            const v2f b1 = *(const v2f*)(b1p + kk);
            const v2f b2 = *(const v2f*)(b2p + kk);
            const v2f b3 = *(const v2f*)(b3p + kk);
            acc0 = __builtin_amdgcn_wmma_f32_16x16x4_f32(false, a, false, b0, (short)0, acc0, false, false);
            acc1 = __builtin_amdgcn_wmma_f32_16x16x4_f32(false, a, false, b1, (short)0, acc1, false, false);
            acc2 = __builtin_amdgcn_wmma_f32_16x16x4_f32(false, a, false, b2, (short)0, acc2, false, false);
            acc3 = __builtin_amdgcn_wmma_f32_16x16x4_f32(false, a, false, b3, (short)0, acc3, false, false);
        }
    } else {
        // B row-major [k][n]: two rows per lane, immediate column offsets
        #pragma unroll 4
        for (int kk = 0; kk < K; kk += 4) {
            const v2f    a  = *(const v2f*)(aRow + kk);
            const float* r0 = B + (long)(kk + 2 * half) * ldb + c;
            const float* r1 = r0 + ldb;
            const v2f b0 = {r0[ 0], r1[ 0]};
            const v2f b1 = {r0[16], r1[16]};
            const v2f b2 = {r0[32], r1[32]};
            const v2f b3 = {r0[48], r1[48]};
            acc0 = __builtin_amdgcn_wmma_f32_16x16x4_f32(false, a, false, b0, (short)0, acc0, false, false);
            acc1 = __builtin_amdgcn_wmma_f32_16x16x4_f32(false, a, false, b1, (short)0, acc1, false, false);
            acc2 = __builtin_amdgcn_wmma_f32_16x16x4_f32(false, a, false, b2, (short)0, acc2, false, false);
            acc3 = __builtin_amdgcn_wmma_f32_16x16x4_f32(false, a, false, b3, (short)0, acc3, false, false);
        }
    }

    // store: D vgpr v -> row m0 + v + 8*half
    float* crow = C + (long)(m0 + 8 * half) * ldc + c;
    #pragma unroll
    for (int v = 0; v < 8; ++v) {
        crow[ 0] = acc0[v];
        crow[16] = acc1[v];
        crow[32] = acc2[v];
        crow[48] = acc3[v];
        crow += ldc;
    }
}

// ---------------------------------------------------------------------------
// out[b] = sum(x[b] * c3), deterministic two-pass reduction.
// Pass 1: 256 chunks x 32 b. Each block reduces 8192 elems (float4 NT loads).
// ---------------------------------------------------------------------------
__global__ __launch_bounds__(256)
void dot_partial(const float* __restrict__ x, const float* __restrict__ c3,
                 float* __restrict__ partial)
{
    const long TOT = 128L * 128 * 128;        // 2097152
    const int  b     = blockIdx.y;
    const int  chunk = blockIdx.x;            // 0..255
    const long base  = (long)chunk * (TOT / 256);

    const f4* xp = (const f4*)(x + b * TOT + base);
    const f4* cp = (const f4*)(c3 + base);

    const int t = threadIdx.x;
    float s = 0.f;
    #pragma unroll 4
    for (int i = t; i < 2048; i += 256) {     // 8192/4 float4 per block
        f4 xv = __builtin_nontemporal_load(xp + i);   // don't pollute L2 (c3 lives there)
        f4 cv = cp[i];
        s += xv.x * cv.x + xv.y * cv.y + xv.z * cv.z + xv.w * cv.w;
    }

    __shared__ float red[256];
    red[t] = s;
    __syncthreads();
    for (int off = 128; off > 0; off >>= 1) {
        if (t < off) red[t] += red[t + off];
        __syncthreads();
    }
    if (t == 0) partial[b * 256 + chunk] = red[0];
}

__global__ __launch_bounds__(256)
void dot_final(const float* __restrict__ partial, float* __restrict__ out)
{
    const int b = blockIdx.x;
    const int t = threadIdx.x;
    __shared__ float red[256];
    red[t] = partial[b * 256 + t];
    __syncthreads();
    for (int off = 128; off > 0; off >>= 1) {
        if (t < off) red[t] += red[t + off];
        __syncthreads();
    }
    if (t == 0) out[b] = red[0];
}

// ---------------------------------------------------------------------------
extern "C" void kernel_launch(void* const* d_in, const int* in_sizes, int n_in,
                              void* d_out, int out_size, void* d_ws, size_t ws_size,
                              hipStream_t stream)
{
    const float* x    = (const float*)d_in[0];   // (32,128,128,128)
    const float* core = (const float*)d_in[1];   // (128,128,128)
    const float* mats = (const float*)d_in[2];   // (3,128,128)
    const float* ops  = (const float*)d_in[3];   // (3,128,2,128,128)
    float* out = (float*)d_out;                  // (32,)
    float* ws  = (float*)d_ws;

    const long T3 = 128L * 128 * 128;            // 2097152
    float* G       = ws;                         // 3*128*128 = 49152 floats
    float* c1      = G + 49152;                  // 2097152
    float* c2      = c1 + T3;
    float* c3      = c2 + T3;
    float* partial = c3 + T3;                    // 32*256

    // v = ops0 @ (ops1 @ mats column)  ->  G0,G1,G2 (row-major 128x128 each)
    front_kernel<<<384, 128, 0, stream>>>(ops, mats, G);

    // mode-0: c1[a, (q,r)] = G0[a,p] * core[p, (q,r)]   (M=128,N=16384,K=128)
    gemm_wmma_f32<false><<<dim3(256, 1, 1), 256, 0, stream>>>(
        G, 128, 0, core, 16384, 0, c1, 16384, 0, 128, 16384, 128);

    // mode-1: per a: c2[a](b,r) = G1[b,q] * c1[a](q,r)  (batch=128, 128^3 each)
    gemm_wmma_f32<false><<<dim3(2, 1, 128), 256, 0, stream>>>(
        G + 16384, 128, 0, c1, 128, T3 / 128, c2, 128, T3 / 128, 128, 128, 128);

    // mode-2: c3[(a,b), c] = c2[(a,b), r] * G2[c,r]^T   (M=16384,N=128,K=128)
    gemm_wmma_f32<true><<<dim3(2, 128, 1), 256, 0, stream>>>(
        c2, 128, 0, G + 32768, 128, 0, c3, 128, 0, 16384, 128, 128);

    // out[b] = <x[b], c3>   (bandwidth-bound: 256 MB of x)
    dot_partial<<<dim3(256, 32), 256, 0, stream>>>(x, c3, partial);
    dot_final<<<32, 256, 0, stream>>>(partial, out);
}